// QuantizerEncoder_75926431858865
// MI455X (gfx1250) — compile-verified
//
#include <hip/hip_runtime.h>
#include <stdint.h>

typedef __attribute__((ext_vector_type(2))) float v2f;
typedef __attribute__((ext_vector_type(8))) float v8f;

#define PQ    32768        // N*H*W query positions
#define MG    4            // codebook groups
#define DD    32           // dims per group
#define KK    2048         // codes per group
#define HWX   4096         // H*W
#define CHWX  (128 * 4096) // C*H*W
#define CHUNK 128          // codes per LDS buffer (16 KB)
#define NCHUNK (KK / CHUNK)
#define TILES  (CHUNK / 16)
#define CHUNK_BYTES (CHUNK * DD * 4)

__global__ __launch_bounds__(256) void vq_argmax_wmma(
    const float* __restrict__ latent,
    const float* __restrict__ codebook,
    int* __restrict__ out)
{
    __shared__ float cb[2][CHUNK * DD];   // double-buffered codebook chunks (32 KB)

    const int m     = blockIdx.y;
    const int tid   = threadIdx.x;
    const int wave  = tid >> 5;
    const int lane  = tid & 31;
    const int laneN = lane & 15;
    const int half  = lane >> 4;

    // each wave handles 32 queries: two 16-row A sub-tiles
    const int p_wave = blockIdx.x * 256 + wave * 32;

    // ---- load A fragments: f32 16x16x4 A layout (lane=M, half selects K pair) ----
    v2f a[2][8];
    #pragma unroll
    for (int s = 0; s < 2; ++s) {
        const int p    = p_wave + s * 16 + laneN;
        const int nimg = p >> 12;          // p / HW
        const int hw   = p & (HWX - 1);    // p % HW
        const float* qbase = latent + (size_t)nimg * CHWX + (size_t)m * DD * HWX + hw;
        #pragma unroll
        for (int ks = 0; ks < 8; ++ks) {
            const int d0 = ks * 4 + half * 2;
            a[s][ks].x = qbase[(size_t)d0 * HWX];
            a[s][ks].y = qbase[(size_t)(d0 + 1) * HWX];
        }
    }

    float bestv[2][8];
    int   besti[2][8];
    #pragma unroll
    for (int s = 0; s < 2; ++s)
        #pragma unroll
        for (int r = 0; r < 8; ++r) { bestv[s][r] = -3.0e38f; besti[s][r] = 0; }

    const uint64_t cbg = (uint64_t)(uintptr_t)(codebook + (size_t)m * KK * DD);
    uint32_t ldsbase[2];
    ldsbase[0] = (uint32_t)(uintptr_t)&cb[0][0];
    ldsbase[1] = (uint32_t)(uintptr_t)&cb[1][0];

    // ---- issue async staging of chunk 0 (256 threads x 4 x b128 = 16 KB) ----
    {
        const uint32_t l0 = ldsbase[0] + tid * 16;
        const uint32_t g0 = tid * 16;
        #pragma unroll
        for (int j = 0; j < 4; ++j) {
            asm volatile("global_load_async_to_lds_b128 %0, %1, %2"
                         :: "v"(l0 + j * 4096), "v"(g0 + j * 4096), "s"(cbg)
                         : "memory");
        }
    }

    for (int ch = 0; ch < NCHUNK; ++ch) {
        const int b = ch & 1;
        // our async writes for chunk ch are done; barrier makes them block-visible
        asm volatile("s_wait_asynccnt 0x0" ::: "memory");
        __syncthreads();

        // prefetch next chunk into the other buffer while we compute on this one
        if (ch + 1 < NCHUNK) {
            const uint32_t l0 = ldsbase[1 - b] + tid * 16;
            const uint32_t g0 = (uint32_t)(ch + 1) * CHUNK_BYTES + tid * 16;
            #pragma unroll
            for (int j = 0; j < 4; ++j) {
                asm volatile("global_load_async_to_lds_b128 %0, %1, %2"
                             :: "v"(l0 + j * 4096), "v"(g0 + j * 4096), "s"(cbg)
                             : "memory");
            }
        }

        #pragma unroll 1
        for (int t = 0; t < TILES; ++t) {
            const int nloc = t * 16 + laneN;
            v8f acc0 = {};
            v8f acc1 = {};
            float ss = 0.0f;   // this lane's half of ||c||^2 (16 of 32 dims)
            #pragma unroll
            for (int ks = 0; ks < 8; ++ks) {
                // B 4x16 layout: lane = N col, half selects K pair
                v2f bf = *(const v2f*)&cb[b][nloc * DD + ks * 4 + half * 2];
                ss = __builtin_fmaf(bf.x, bf.x, ss);
                ss = __builtin_fmaf(bf.y, bf.y, ss);
                acc0 = __builtin_amdgcn_wmma_f32_16x16x4_f32(
                    false, a[0][ks], false, bf, (short)0, acc0, false, false);
                acc1 = __builtin_amdgcn_wmma_f32_16x16x4_f32(
                    false, a[1][ks], false, bf, (short)0, acc1, false, false);
            }
            // halves hold disjoint dim sets of the same code -> full norm via xor16
            const float c2v  = ss + __shfl_xor(ss, 16, 32);
            const int   kidx = ch * CHUNK + t * 16 + laneN;
            #pragma unroll
            for (int r = 0; r < 8; ++r) {
                float s0 = 2.0f * acc0[r] - c2v;
                if (s0 > bestv[0][r]) { bestv[0][r] = s0; besti[0][r] = kidx; }
                float s1 = 2.0f * acc1[r] - c2v;
                if (s1 > bestv[1][r]) { bestv[1][r] = s1; besti[1][r] = kidx; }
            }
        }
        // no trailing barrier needed: next iteration's wait+barrier precedes any
        // async refill of this buffer
    }

    // ---- cross-lane argmax within each 16-lane half (xor masks <= 8 stay in half) ----
    #pragma unroll
    for (int s = 0; s < 2; ++s) {
        #pragma unroll
        for (int r = 0; r < 8; ++r) {
            float v = bestv[s][r];
            int   i = besti[s][r];
            #pragma unroll
            for (int off = 8; off >= 1; off >>= 1) {
                float v2 = __shfl_xor(v, off, 32);
                int   i2 = __shfl_xor(i, off, 32);
                if (v2 > v || (v2 == v && i2 < i)) { v = v2; i = i2; }
            }
            bestv[s][r] = v; besti[s][r] = i;
        }
    }

    // C/D layout: lanes 0-15 hold rows M=r, lanes 16-31 hold rows M=8+r
    #pragma unroll
    for (int s = 0; s < 2; ++s) {
        #pragma unroll
        for (int r = 0; r < 8; ++r) {
            if (laneN == r) {
                const int row = p_wave + s * 16 + r + 8 * half;
                out[(size_t)row * MG + m] = besti[s][r];
            }
        }
    }
}

extern "C" void kernel_launch(void* const* d_in, const int* in_sizes, int n_in,
                              void* d_out, int out_size, void* d_ws, size_t ws_size,
                              hipStream_t stream) {
    (void)in_sizes; (void)n_in; (void)out_size; (void)d_ws; (void)ws_size;
    const float* latent   = (const float*)d_in[0];
    const float* codebook = (const float*)d_in[1];
    int* out = (int*)d_out;
    dim3 grid(PQ / 256, MG);
    vq_argmax_wmma<<<grid, 256, 0, stream>>>(latent, codebook, out);
}